// Decoder_34127810134343
// MI455X (gfx1250) — compile-verified
//
#include <hip/hip_runtime.h>

// Problem constants (match reference)
#define Bsz 8
#define LAT 512
#define NF 64
#define Hdim 128
#define HW (Hdim * Hdim)        // 16384
#define KC (3 * NF)             // 192
#define NCALLS 32
#define EPSV 1e-5f

typedef float v2f __attribute__((ext_vector_type(2)));
typedef float v8f __attribute__((ext_vector_type(8)));

// ---------------------------------------------------------------------------
// Kernel 0: Wd = lat @ w_dyn + b_dyn  (8x12288), bd = lat @ w_bias + b_bias (8x64)
// ---------------------------------------------------------------------------
__global__ __launch_bounds__(256)
void dyn_kernel(const float* __restrict__ lat, const float* __restrict__ w_dyn,
                const float* __restrict__ b_dyn, const float* __restrict__ w_bias,
                const float* __restrict__ b_bias, float* __restrict__ Wd,
                float* __restrict__ bd) {
  int idx = blockIdx.x * 256 + threadIdx.x;
  const int NWD = Bsz * NF * KC;     // 98304
  if (idx < NWD) {
    int b = idx / (NF * KC);
    int j = idx % (NF * KC);         // j = o*KC + c  (row-major reshape)
    const float* l = lat + b * LAT;
    float acc = b_dyn[j];
    for (int k = 0; k < LAT; ++k) acc += l[k] * w_dyn[(size_t)k * (NF * KC) + j];
    Wd[idx] = acc;
  } else if (idx < NWD + Bsz * NF) {
    int i = idx - NWD;
    int b = i / NF, o = i % NF;
    const float* l = lat + b * LAT;
    float acc = b_bias[o];
    for (int k = 0; k < LAT; ++k) acc += l[k] * w_bias[k * NF + o];
    bd[i] = acc;
  }
}

// ---------------------------------------------------------------------------
// Kernel 1: pre-life mask from maxpool3(state ch3) > 0.1 ; also zero stats buf
// ---------------------------------------------------------------------------
__device__ __forceinline__ float maxpool3_at(const float* __restrict__ plane,
                                             int y, int x) {
  float m = -3.4e38f;
  for (int dy = -1; dy <= 1; ++dy) {
    int yy = y + dy; if (yy < 0 || yy >= Hdim) continue;
    for (int dx = -1; dx <= 1; ++dx) {
      int xx = x + dx; if (xx < 0 || xx >= Hdim) continue;
      float v = plane[yy * Hdim + xx];
      m = fmaxf(m, v);
    }
  }
  return m;
}

__global__ __launch_bounds__(256)
void prelife_kernel(const float* __restrict__ S, float* __restrict__ mask,
                    float* __restrict__ sums) {
  int gid = blockIdx.x * 256 + threadIdx.x;
  if (gid < 2 * Bsz * KC) sums[gid] = 0.0f;   // zero stats for this step
  int b = gid >> 14;
  int p = gid & (HW - 1);
  const float* plane = S + ((size_t)b * NF + 3) * HW;
  float m = maxpool3_at(plane, p >> 7, p & 127);
  mask[gid] = (m > 0.1f) ? 1.0f : 0.0f;
}

// ---------------------------------------------------------------------------
// Kernel 2: depthwise sobel (ident, sx, sy) + per-channel sum/sumsq stats
// grid: Bsz * NF * (HW/256) blocks; block handles one (b,c) pixel chunk
// ---------------------------------------------------------------------------
__global__ __launch_bounds__(256)
void sobel_stats_kernel(const float* __restrict__ S, float* __restrict__ Y,
                        float* __restrict__ sums) {
  int blk = blockIdx.x;
  int chunk = blk & 63;
  int c = (blk >> 6) & (NF - 1);
  int b = blk >> 12;
  int p = chunk * 256 + threadIdx.x;
  int y = p >> 7, x = p & 127;

  const float* Sp = S + ((size_t)b * NF + c) * HW;
  float v[3][3];
  for (int dy = -1; dy <= 1; ++dy)
    for (int dx = -1; dx <= 1; ++dx) {
      int yy = y + dy, xx = x + dx;
      v[dy + 1][dx + 1] = (yy >= 0 && yy < Hdim && xx >= 0 && xx < Hdim)
                              ? Sp[yy * Hdim + xx] : 0.0f;  // SAME zero pad
    }
  float id = v[1][1];
  float sx = (-v[0][0] + v[0][2] - 2.f * v[1][0] + 2.f * v[1][2]
              - v[2][0] + v[2][2]) * 0.125f;
  float sy = (-v[0][0] - 2.f * v[0][1] - v[0][2]
              + v[2][0] + 2.f * v[2][1] + v[2][2]) * 0.125f;

  float* Yb = Y + ((size_t)b * KC + 3 * c) * HW + p;
  Yb[0]                 = id;
  Yb[(size_t)HW]        = sx;
  Yb[(size_t)2 * HW]    = sy;

  // block reduction of 6 stats -> atomics
  __shared__ float sd[256];
  float vals[6] = {id, id * id, sx, sx * sx, sy, sy * sy};
  int tid = threadIdx.x;
  for (int q = 0; q < 6; ++q) {
    sd[tid] = vals[q];
    __syncthreads();
    for (int s = 128; s > 0; s >>= 1) {
      if (tid < s) sd[tid] += sd[tid + s];
      __syncthreads();
    }
    if (tid == 0)
      atomicAdd(&sums[((size_t)b * KC + 3 * c + (q >> 1)) * 2 + (q & 1)], sd[0]);
    __syncthreads();
  }
}

// ---------------------------------------------------------------------------
// Kernel 3: fold instance-norm into weights: W' = Wd*r_c, b' = bd - sum Wd*r*mu
// one thread per (b,o)
// ---------------------------------------------------------------------------
__global__ __launch_bounds__(64)
void finalize_kernel(const float* __restrict__ Wd, const float* __restrict__ bd,
                     const float* __restrict__ sums, float* __restrict__ Wp,
                     float* __restrict__ bp) {
  int b = blockIdx.x, o = threadIdx.x;
  const float invHW = 1.0f / (float)HW;
  const float* wrow = Wd + ((size_t)b * NF + o) * KC;
  float* prow = Wp + ((size_t)b * NF + o) * KC;
  float acc = 0.0f;
  for (int c = 0; c < KC; ++c) {
    float s  = sums[((size_t)b * KC + c) * 2 + 0];
    float s2 = sums[((size_t)b * KC + c) * 2 + 1];
    float mu = s * invHW;
    float var = s2 * invHW - mu * mu;
    float r = rsqrtf(var + EPSV);
    float wp = wrow[c] * r;
    prow[c] = wp;
    acc += wp * mu;
  }
  bp[b * NF + o] = bd[b * NF + o] - acc;
}

// ---------------------------------------------------------------------------
// Kernel 4: WMMA GEMM  Out[b][64][HW] = W'[b][64][192] x Y[b][192][HW] + b'
// Block: 256 thr (8 waves). Wave w: m-tile = w&3 (16 rows), n-group = w>>2
// (64 cols). Block covers M=64 x N=128 (one image row).
// A (W') staged in LDS with pad to avoid bank conflicts.
// ---------------------------------------------------------------------------
#define APAD 196   // 192 + 4 (keeps 8B alignment, banks spread)
__global__ __launch_bounds__(256)
void gemm_kernel(const float* __restrict__ Y, const float* __restrict__ Wp,
                 const float* __restrict__ bp, float* __restrict__ Out) {
  __shared__ float lA[NF * APAD];
  int b = blockIdx.x >> 7;
  int nblk = blockIdx.x & 127;

  const float* Wb = Wp + (size_t)b * NF * KC;
  for (int i = threadIdx.x; i < NF * KC; i += 256) {
    int r = i / KC, cc = i % KC;
    lA[r * APAD + cc] = Wb[i];
  }
  __syncthreads();

  int lane = threadIdx.x & 31;
  int wave = threadIdx.x >> 5;
  int mt = wave & 3;          // M tile (16 rows)
  int ng = wave >> 2;         // 0..1 -> 64 cols each
  int half = lane >> 4;       // 0/1 selects K pair / M half
  int l16 = lane & 15;
  int col0 = nblk * 128 + ng * 64;

  const float* Yb = Y + (size_t)b * KC * HW;
  const float* Arow = &lA[(mt * 16 + l16) * APAD];

  v8f acc[4] = {};
  for (int k = 0; k < KC; k += 4) {
    int ka = k + half * 2;
    v2f a;
    a.x = Arow[ka];
    a.y = Arow[ka + 1];
    const float* Yk = Yb + (size_t)ka * HW;
#pragma unroll
    for (int t = 0; t < 4; ++t) {
      int col = col0 + t * 16 + l16;
      v2f bf;
      bf.x = Yk[col];               // row ka
      bf.y = Yk[(size_t)HW + col];  // row ka+1
      acc[t] = __builtin_amdgcn_wmma_f32_16x16x4_f32(
          false, a, false, bf, (short)0, acc[t], false, false);
    }
  }

  // epilogue: + bias, store. D layout: VGPR j -> row mt*16 + half*8 + j
  const float* bpb = bp + b * NF;
  float bias_j[8];
#pragma unroll
  for (int j = 0; j < 8; ++j) bias_j[j] = bpb[mt * 16 + half * 8 + j];
  float* Ob = Out + (size_t)b * NF * HW;
#pragma unroll
  for (int t = 0; t < 4; ++t) {
    int col = col0 + t * 16 + l16;
#pragma unroll
    for (int j = 0; j < 8; ++j) {
      int m = mt * 16 + half * 8 + j;
      Ob[(size_t)m * HW + col] = acc[t][j] + bias_j[j];
    }
  }
}

// ---------------------------------------------------------------------------
// Kernel 5: post-life maxpool + life/leak scaling -> next state
// ---------------------------------------------------------------------------
__global__ __launch_bounds__(256)
void maskscale_kernel(const float* __restrict__ Yn, const float* __restrict__ mask,
                      const float* __restrict__ leak_p, float* __restrict__ Snext) {
  int gid = blockIdx.x * 256 + threadIdx.x;
  int b = gid >> 14;
  int p = gid & (HW - 1);
  float leak = fminf(fmaxf(leak_p[0], 0.001f), 1000.0f);
  const float* plane = Yn + ((size_t)b * NF + 3) * HW;
  float post = maxpool3_at(plane, p >> 7, p & 127);
  float life = ((mask[gid] != 0.0f) && (post > 0.1f)) ? 1.0f : 0.0f;
  float factor = life + leak;
  const float* src = Yn + (size_t)b * NF * HW + p;
  float* dst = Snext + (size_t)b * NF * HW + p;
  for (int o = 0; o < NF; ++o)
    dst[(size_t)o * HW] = src[(size_t)o * HW] * factor;
}

// ---------------------------------------------------------------------------
// Kernel 6: final clip of channels 0..2
// ---------------------------------------------------------------------------
__global__ __launch_bounds__(256)
void out_kernel(const float* __restrict__ S, float* __restrict__ out) {
  int idx = blockIdx.x * 256 + threadIdx.x;
  int b = idx / (3 * HW);
  int c = (idx / HW) % 3;
  int p = idx & (HW - 1);
  float v = S[((size_t)b * NF + c) * HW + p];
  out[idx] = fminf(fmaxf(v, -1.0f), 1.0f);
}

// ---------------------------------------------------------------------------
extern "C" void kernel_launch(void* const* d_in, const int* in_sizes, int n_in,
                              void* d_out, int out_size, void* d_ws, size_t ws_size,
                              hipStream_t stream) {
  const float* lat     = (const float*)d_in[0];
  const float* ca_init = (const float*)d_in[1];
  const float* w_dyn   = (const float*)d_in[2];
  const float* b_dyn   = (const float*)d_in[3];
  const float* w_bias  = (const float*)d_in[4];
  const float* b_bias  = (const float*)d_in[5];
  const float* leak    = (const float*)d_in[6];
  float* out = (float*)d_out;

  // workspace layout (floats)
  float* ws = (float*)d_ws;
  const size_t N_STATE = (size_t)Bsz * NF * HW;   // 8.39M
  const size_t N_Y     = (size_t)Bsz * KC * HW;   // 25.2M
  float* stateA = ws;                 ws += N_STATE;
  float* stateB = ws;                 ws += N_STATE;
  float* Ybuf   = ws;                 ws += N_Y;
  float* sums   = ws;                 ws += (size_t)2 * Bsz * KC;
  float* Wd     = ws;                 ws += (size_t)Bsz * NF * KC;
  float* bd     = ws;                 ws += (size_t)Bsz * NF;
  float* Wp     = ws;                 ws += (size_t)Bsz * NF * KC;
  float* bp     = ws;                 ws += (size_t)Bsz * NF;
  float* maskb  = ws;                 ws += (size_t)Bsz * HW;

  // 0) dynamic weights
  {
    int total = Bsz * NF * KC + Bsz * NF;
    dyn_kernel<<<(total + 255) / 256, 256, 0, stream>>>(lat, w_dyn, b_dyn,
                                                        w_bias, b_bias, Wd, bd);
  }
  // init state
  hipMemcpyAsync(stateA, ca_init, N_STATE * sizeof(float),
                 hipMemcpyDeviceToDevice, stream);

  for (int step = 0; step < NCALLS; ++step) {
    prelife_kernel<<<Bsz * HW / 256, 256, 0, stream>>>(stateA, maskb, sums);
    sobel_stats_kernel<<<Bsz * NF * (HW / 256), 256, 0, stream>>>(stateA, Ybuf,
                                                                  sums);
    finalize_kernel<<<Bsz, NF, 0, stream>>>(Wd, bd, sums, Wp, bp);
    gemm_kernel<<<Bsz * (HW / 128), 256, 0, stream>>>(Ybuf, Wp, bp, stateB);
    maskscale_kernel<<<Bsz * HW / 256, 256, 0, stream>>>(stateB, maskb, leak,
                                                         stateA);
  }

  out_kernel<<<(Bsz * 3 * HW) / 256, 256, 0, stream>>>(stateA, out);
}